// Dihedral_20547123544381
// MI455X (gfx1250) — compile-verified
//
#include <hip/hip_runtime.h>

// Problem constants (match reference)
#define N_ATOMS 200000
#define N_DIH   2000000
#define N_TYPES 25
#define N_DEGS  5
#define N_MOL   1024
#define TBL_STRIDE (N_TYPES * N_TYPES * N_TYPES * N_TYPES)  // 390625 per degree slice

__global__ void zero_energy_kernel(float* __restrict__ out) {
    int i = blockIdx.x * blockDim.x + threadIdx.x;
    if (i < N_MOL) out[i] = 0.0f;
}

__global__ __launch_bounds__(256) void dihedral_energy_kernel(
    const float* __restrict__ pos,            // (N_ATOMS, 3)
    const float* __restrict__ k1s,            // (N_DEGS, 25,25,25,25)
    const float* __restrict__ k2s,            // (N_DEGS, 25,25,25,25)
    const int*   __restrict__ mapping,        // (4, N_DIH)
    const int*   __restrict__ atom_types,     // (N_ATOMS,)
    const int*   __restrict__ mapping_batch,  // (N_DIH,) sorted
    float*       __restrict__ out)            // (N_MOL,)
{
    const int gid   = blockIdx.x * blockDim.x + threadIdx.x;
    const bool valid = (gid < N_DIH);
    const int i = valid ? gid : (N_DIH - 1);   // clamp for safe loads; result masked

    // ---- coalesced index / batch loads -------------------------------------
    const int i0 = mapping[0 * N_DIH + i];
    const int i1 = mapping[1 * N_DIH + i];
    const int i2 = mapping[2 * N_DIH + i];
    const int i3 = mapping[3 * N_DIH + i];
    const int b  = valid ? mapping_batch[i] : -1;

    // ---- position gathers (random, L2-resident: pos = 2.4 MB) --------------
    const float p0x = pos[3 * i0 + 0], p0y = pos[3 * i0 + 1], p0z = pos[3 * i0 + 2];
    const float p1x = pos[3 * i1 + 0], p1y = pos[3 * i1 + 1], p1z = pos[3 * i1 + 2];
    const float p2x = pos[3 * i2 + 0], p2y = pos[3 * i2 + 1], p2z = pos[3 * i2 + 2];
    const float p3x = pos[3 * i3 + 0], p3y = pos[3 * i3 + 1], p3z = pos[3 * i3 + 2];

    // ---- bond vectors -------------------------------------------------------
    const float b1x = p1x - p0x, b1y = p1y - p0y, b1z = p1z - p0z;
    const float b2x = p2x - p1x, b2y = p2y - p1y, b2z = p2z - p1z;
    const float b3x = p3x - p2x, b3y = p3y - p2y, b3z = p3z - p2z;

    // n1 = b1 x b2 ; n2 = b2 x b3 ; mm = n1 x b2
    const float n1x = b1y * b2z - b1z * b2y;
    const float n1y = b1z * b2x - b1x * b2z;
    const float n1z = b1x * b2y - b1y * b2x;

    const float n2x = b2y * b3z - b2z * b3y;
    const float n2y = b2z * b3x - b2x * b3z;
    const float n2z = b2x * b3y - b2y * b3x;

    const float mmx = n1y * b2z - n1z * b2y;
    const float mmy = n1z * b2x - n1x * b2z;
    const float mmz = n1x * b2y - n1y * b2x;

    // x = n1.n2 ; y = (n1 x b2_unit).n2 = (mm.n2) / |b2|
    const float x   = n1x * n2x + n1y * n2y + n1z * n2z;
    const float ymm = mmx * n2x + mmy * n2y + mmz * n2z;
    const float b2l2 = b2x * b2x + b2y * b2y + b2z * b2z;
    const float y = ymm * rsqrtf(fmaxf(b2l2, 1e-30f));

    // sin/cos of theta = atan2(y, x) without any transcendental beyond rsqrt
    const float r2   = x * x + y * y;
    const float rinv = (r2 > 0.0f) ? rsqrtf(r2) : 0.0f;
    const float cth  = x * rinv;   // cos(theta)
    const float sth  = y * rinv;   // sin(theta)

    // ---- type tuple -> table index -----------------------------------------
    const int t0 = atom_types[i0];
    const int t1 = atom_types[i1];
    const int t2 = atom_types[i2];
    const int t3 = atom_types[i3];
    const int tidx = ((t0 * N_TYPES + t1) * N_TYPES + t2) * N_TYPES + t3;

    // ---- Fourier series: V = sum_d k1[d]*sin(d*th) + k2[d]*cos(d*th) -------
    // d = 0: sin term is 0, cos term is k2s[idx]
    float V  = k2s[tidx];
    float sd = sth;               // sin(1*theta)
    float cd = cth;               // cos(1*theta)
#pragma unroll
    for (int d = 1; d < N_DEGS; ++d) {
        V = fmaf(k1s[d * TBL_STRIDE + tidx], sd, V);
        V = fmaf(k2s[d * TBL_STRIDE + tidx], cd, V);
        // angle addition: (s,c)_{d+1} = (s_d*c1 + c_d*s1, c_d*c1 - s_d*s1)
        const float sn = fmaf(sd, cth,  cd * sth);
        const float cn = fmaf(cd, cth, -sd * sth);
        sd = sn;
        cd = cn;
    }
    float Vl = valid ? V : 0.0f;

    // ---- wave32 segmented reduction over sorted batch ids -------------------
    // Lanes holding the same batch id are contiguous (batch sorted). Each lane
    // accumulates the suffix of its segment within the wave; segment heads do
    // one global atomic. Avg segment ~1953 -> ~1 atomic per wave.
    const int lane = threadIdx.x & 31;  // wave32
#pragma unroll
    for (int off = 1; off < 32; off <<= 1) {
        const float vo = __shfl_down(Vl, off, 32);
        const int   bo = __shfl_down(b,  off, 32);
        if ((lane + off) < 32 && bo == b) Vl += vo;
    }
    const int bprev = __shfl_up(b, 1, 32);
    const bool head = (lane == 0) || (bprev != b);
    if (valid && head) {
        atomicAdd(&out[b], Vl);
    }
}

extern "C" void kernel_launch(void* const* d_in, const int* in_sizes, int n_in,
                              void* d_out, int out_size, void* d_ws, size_t ws_size,
                              hipStream_t stream) {
    (void)in_sizes; (void)n_in; (void)out_size; (void)d_ws; (void)ws_size;

    const float* pos           = (const float*)d_in[0];
    const float* k1s           = (const float*)d_in[1];
    const float* k2s           = (const float*)d_in[2];
    const int*   mapping       = (const int*)d_in[3];
    const int*   atom_types    = (const int*)d_in[4];
    const int*   mapping_batch = (const int*)d_in[5];
    float*       out           = (float*)d_out;

    // d_out is poisoned by the harness; zero it every call (deterministic).
    zero_energy_kernel<<<(N_MOL + 255) / 256, 256, 0, stream>>>(out);

    const int block = 256;
    const int grid  = (N_DIH + block - 1) / block;
    dihedral_energy_kernel<<<grid, block, 0, stream>>>(
        pos, k1s, k2s, mapping, atom_types, mapping_batch, out);
}